// BidiagonalPCRFactorization_35210141892962
// MI455X (gfx1250) — compile-verified
//
#include <hip/hip_runtime.h>

// Block lower-bidiagonal solve  x_i = A_i^{-1}(v_i - B_{i-1} x_{i-1})
// via affine-recurrence chunked parallel scan with homogeneous 16x16
// composition done in V_WMMA_F32_16X16X4_F32.
//
// Workspace layout (floats), requires ws_size >= ~85 MB:
//   Mw : [NBLK][SBAT][8][8]   recurrence matrices  M_i = -A_i^{-1} B_{i-1}
//   cw : [NBLK][SBAT][8]      recurrence offsets   c_i =  A_i^{-1} v_i
//   Gw : [NCH ][SBAT][16][16] per-chunk composed homogeneous matrices
//   Ew : [NCH ][SBAT][8]      x entering each chunk

#define NBLK 1024
#define SBAT 256
#define SBLK 8
#define LCH  32
#define NCH  (NBLK / LCH)

typedef float v2f __attribute__((ext_vector_type(2)));
typedef float v8f __attribute__((ext_vector_type(8)));

// ---------------------------------------------------------------------------
// K1: per (i,b) -> invert A (Gauss-Jordan, no pivoting: diagonally dominant),
//     M = -Ainv*Bprev (M_0 = 0), c = Ainv*v_i.  One thread per 8x8 system.
// ---------------------------------------------------------------------------
__global__ void k1_precompute(const float* __restrict__ A,
                              const float* __restrict__ B,
                              const float* __restrict__ v,
                              float* __restrict__ Mw,
                              float* __restrict__ cw) {
  const int T = blockIdx.x * blockDim.x + threadIdx.x;   // i*SBAT + b
  const int i = T >> 8;
  const int b = T & (SBAT - 1);

  const float* __restrict__ Ai = A + (size_t)T * 64;
  float a[64], inv[64];
#pragma unroll
  for (int k = 0; k < 16; ++k) {
    float4 t4 = ((const float4*)Ai)[k];
    a[4 * k + 0] = t4.x; a[4 * k + 1] = t4.y;
    a[4 * k + 2] = t4.z; a[4 * k + 3] = t4.w;
  }
#pragma unroll
  for (int r = 0; r < 8; ++r)
#pragma unroll
    for (int c = 0; c < 8; ++c) inv[r * 8 + c] = (r == c) ? 1.f : 0.f;

#pragma unroll
  for (int p = 0; p < 8; ++p) {
    const float pr = 1.f / a[p * 8 + p];
#pragma unroll
    for (int c = 0; c < 8; ++c) { a[p * 8 + c] *= pr; inv[p * 8 + c] *= pr; }
#pragma unroll
    for (int r = 0; r < 8; ++r) {
      if (r != p) {
        const float f = a[r * 8 + p];
#pragma unroll
        for (int c = 0; c < 8; ++c) {
          a[r * 8 + c]   -= f * a[p * 8 + c];
          inv[r * 8 + c] -= f * inv[p * 8 + c];
        }
      }
    }
  }

  // c_i = inv * v_i
  const float* __restrict__ vi = v + (size_t)b * (NBLK * SBLK) + (size_t)i * SBLK;
  float vv[8];
#pragma unroll
  for (int k = 0; k < 8; ++k) vv[k] = vi[k];
  float* __restrict__ co = cw + (size_t)T * 8;
#pragma unroll
  for (int r = 0; r < 8; ++r) {
    float s = 0.f;
#pragma unroll
    for (int k = 0; k < 8; ++k) s += inv[r * 8 + k] * vv[k];
    co[r] = s;
  }

  // M_i = -inv * B_{i-1}   (M_0 = 0)
  float* __restrict__ Mo = Mw + (size_t)T * 64;
  if (i == 0) {
#pragma unroll
    for (int k = 0; k < 64; ++k) Mo[k] = 0.f;
  } else {
    const float* __restrict__ Bi = B + ((size_t)(i - 1) * SBAT + b) * 64;
    float bb[64];
#pragma unroll
    for (int k = 0; k < 16; ++k) {
      float4 t4 = ((const float4*)Bi)[k];
      bb[4 * k + 0] = t4.x; bb[4 * k + 1] = t4.y;
      bb[4 * k + 2] = t4.z; bb[4 * k + 3] = t4.w;
    }
#pragma unroll
    for (int r = 0; r < 8; ++r)
#pragma unroll
      for (int c = 0; c < 8; ++c) {
        float s = 0.f;
#pragma unroll
        for (int k = 0; k < 8; ++k) s += inv[r * 8 + k] * bb[k * 8 + c];
        Mo[r * 8 + c] = -s;
      }
  }
}

// ---------------------------------------------------------------------------
// K2a: one wave per (chunk g, batch b). Compose H_t = [[M,c,0],[0,1,0],[0,0,I]]
//      over the chunk:  P <- H_t * P  with 4x V_WMMA_F32_16X16X4_F32 per step.
//      A-operand layout (ISA): lane l -> row r=l%16, K-pair (2*hi, 2*hi+1)+4q.
//      C/D layout (ISA): VGPR r -> lanes0-15 row r, lanes16-31 row r+8.
//      B-operand layout assumed to mirror A (rows k striped like A's K).
// ---------------------------------------------------------------------------
__global__ void k2a_compose(const float* __restrict__ Mw,
                            const float* __restrict__ cw,
                            float* __restrict__ Gw) {
  const int lane = threadIdx.x & 31;
  const int task = blockIdx.x * (blockDim.x >> 5) + (threadIdx.x >> 5);
  const int g = task >> 8;           // / SBAT
  const int b = task & (SBAT - 1);
  const int r  = lane & 15;          // row (A/C layouts) == col n (B/C layouts)
  const int hi = lane >> 4;

  // P = I in B-operand layout: bm[q][v] holds P[4q+v+2hi, n=r]
  float bm[4][2];
#pragma unroll
  for (int q = 0; q < 4; ++q)
#pragma unroll
    for (int v = 0; v < 2; ++v)
      bm[q][v] = ((4 * q + v + 2 * hi) == r) ? 1.f : 0.f;

  for (int t = g * LCH; t < g * LCH + LCH; ++t) {
    const size_t mbase = ((size_t)t * SBAT + b) * 64;
    float m0 = 0.f, m1 = 0.f, m4 = 0.f, m5 = 0.f, cv = 0.f;
    if (r < 8) {
      const float* __restrict__ Mrow = Mw + mbase + r * 8 + 2 * hi;
      m0 = Mrow[0]; m1 = Mrow[1]; m4 = Mrow[4]; m5 = Mrow[5];
      if (hi == 0) cv = cw[((size_t)t * SBAT + b) * 8 + r];
    }
    // Build H_t in A-operand layout (k-chunks q=0..3)
    float a20, a21, a30, a31;
    if (r < 8) {
      a20 = (hi == 0) ? cv : 0.f;               // H[r,8] = c_r
      a21 = 0.f; a30 = 0.f; a31 = 0.f;
    } else {                                    // identity padding rows
      a20 = ((8  + 2 * hi) == r) ? 1.f : 0.f;
      a21 = ((9  + 2 * hi) == r) ? 1.f : 0.f;
      a30 = ((12 + 2 * hi) == r) ? 1.f : 0.f;
      a31 = ((13 + 2 * hi) == r) ? 1.f : 0.f;
    }
    v2f A0 = {m0, m1};          // k = 2hi, 2hi+1          (M cols 0..3)
    v2f A1 = {m4, m5};          // k = 4+2hi, 5+2hi        (M cols 4..7)
    v2f A2 = {a20, a21};        // k = 8..11 : c column + identity
    v2f A3 = {a30, a31};        // k = 12..15: identity
    v2f B0 = {bm[0][0], bm[0][1]};
    v2f B1 = {bm[1][0], bm[1][1]};
    v2f B2 = {bm[2][0], bm[2][1]};
    v2f B3 = {bm[3][0], bm[3][1]};

    v8f acc = {0.f, 0.f, 0.f, 0.f, 0.f, 0.f, 0.f, 0.f};
    acc = __builtin_amdgcn_wmma_f32_16x16x4_f32(false, A0, false, B0, (short)0, acc, false, false);
    acc = __builtin_amdgcn_wmma_f32_16x16x4_f32(false, A1, false, B1, (short)0, acc, false, false);
    acc = __builtin_amdgcn_wmma_f32_16x16x4_f32(false, A2, false, B2, (short)0, acc, false, false);
    acc = __builtin_amdgcn_wmma_f32_16x16x4_f32(false, A3, false, B3, (short)0, acc, false, false);

    // D (C-layout) -> P (B-operand layout) for next iteration.
    // Rows k<8 live in VGPR k lanes0-15; rows k>=8 in VGPR k-8 lanes16-31.
#pragma unroll
    for (int q = 0; q < 2; ++q)                  // target rows k0=4q+v (<8), k1=k0+2 (<8)
#pragma unroll
      for (int v = 0; v < 2; ++v) {
        const float lo = acc[4 * q + v];                       // hi=0: local
        const float ot = __shfl_xor(acc[4 * q + v + 2], 16, 32); // hi=1: from lane-16
        bm[q][v] = hi ? ot : lo;
      }
#pragma unroll
    for (int q = 2; q < 4; ++q)                  // target rows >= 8
#pragma unroll
      for (int v = 0; v < 2; ++v) {
        const float hv = acc[4 * q + v + 2 - 8];               // hi=1: local
        const float ot = __shfl_xor(acc[4 * q + v - 8], 16, 32); // hi=0: from lane+16
        bm[q][v] = hi ? hv : ot;
      }
  }

  // Store composed chunk matrix [16][16] row-major from B-operand layout.
  float* __restrict__ Go = Gw + ((size_t)g * SBAT + b) * 256;
#pragma unroll
  for (int q = 0; q < 4; ++q)
#pragma unroll
    for (int v = 0; v < 2; ++v) {
      const int k = 4 * q + v + 2 * hi;
      Go[k * 16 + r] = bm[q][v];
    }
}

// ---------------------------------------------------------------------------
// K2b: sequential scan over 32 chunks; 8 lanes per batch.
//      Ew[g][b] = x entering chunk g; x <- P_g x + s_g.
// ---------------------------------------------------------------------------
__global__ void k2b_scan(const float* __restrict__ Gw, float* __restrict__ Ew) {
  const int T = blockIdx.x * blockDim.x + threadIdx.x;   // 0..SBAT*8-1
  const int b = T >> 3;
  const int rr = T & 7;
  const int lanebase = (threadIdx.x & 31) & ~7;

  float x = 0.f;                                          // x[rr]
  for (int g = 0; g < NCH; ++g) {
    Ew[((size_t)g * SBAT + b) * 8 + rr] = x;
    const size_t base = ((size_t)g * SBAT + b) * 256 + (size_t)rr * 16;
    float Prow[8];
#pragma unroll
    for (int k = 0; k < 8; ++k) Prow[k] = Gw[base + k];
    const float s = Gw[base + 8];
    float xn = s;
#pragma unroll
    for (int k = 0; k < 8; ++k) {
      const float xk = __shfl(x, lanebase + k, 32);
      xn += Prow[k] * xk;
    }
    x = xn;
  }
}

// ---------------------------------------------------------------------------
// K2c: one thread per (chunk g, batch b); apply x <- M_t x + c_t within chunk,
//      writing each x_t to the output in (sbat, nblk*sblk) layout.
// ---------------------------------------------------------------------------
__global__ void k2c_apply(const float* __restrict__ Mw,
                          const float* __restrict__ cw,
                          const float* __restrict__ Ew,
                          float* __restrict__ out) {
  const int T = blockIdx.x * blockDim.x + threadIdx.x;   // 0..NCH*SBAT-1
  const int g = T >> 8;
  const int b = T & (SBAT - 1);

  float x[8];
#pragma unroll
  for (int j = 0; j < 8; ++j) x[j] = Ew[((size_t)g * SBAT + b) * 8 + j];

  for (int t = g * LCH; t < g * LCH + LCH; ++t) {
    const size_t mb = ((size_t)t * SBAT + b) * 64;
    const size_t cb = ((size_t)t * SBAT + b) * 8;
    float xn[8];
#pragma unroll
    for (int rr = 0; rr < 8; ++rr) {
      float s = cw[cb + rr];
#pragma unroll
      for (int k = 0; k < 8; ++k) s += Mw[mb + rr * 8 + k] * x[k];
      xn[rr] = s;
    }
    float* __restrict__ op = out + (size_t)b * (NBLK * SBLK) + (size_t)t * 8;
#pragma unroll
    for (int j = 0; j < 8; ++j) { x[j] = xn[j]; op[j] = x[j]; }
  }
}

// ---------------------------------------------------------------------------
extern "C" void kernel_launch(void* const* d_in, const int* in_sizes, int n_in,
                              void* d_out, int out_size, void* d_ws, size_t ws_size,
                              hipStream_t stream) {
  (void)in_sizes; (void)n_in; (void)out_size; (void)ws_size;
  const float* A = (const float*)d_in[0];   // (1024, 256, 8, 8)
  const float* B = (const float*)d_in[1];   // (1023, 256, 8, 8)
  const float* v = (const float*)d_in[2];   // (256, 8192)
  float* out = (float*)d_out;               // (256, 8192)

  float* Mw = (float*)d_ws;
  float* cw = Mw + (size_t)NBLK * SBAT * 64;
  float* Gw = cw + (size_t)NBLK * SBAT * 8;
  float* Ew = Gw + (size_t)NCH * SBAT * 256;   // total ~85 MB of ws

  k1_precompute<<<(NBLK * SBAT) / 256, 256, 0, stream>>>(A, B, v, Mw, cw);
  k2a_compose<<<(NCH * SBAT) / 8, 256, 0, stream>>>(Mw, cw, Gw);
  k2b_scan<<<(SBAT * 8) / 256, 256, 0, stream>>>(Gw, Ew);
  k2c_apply<<<(NCH * SBAT) / 256, 256, 0, stream>>>(Mw, cw, Ew, out);
}